// normalized_P_SDMA_21827023798275
// MI455X (gfx1250) — compile-verified
//
#include <hip/hip_runtime.h>

// MI455X / gfx1250, wave32. Two-pass bandwidth-bound normalization:
//   pass 1: per-antenna energy reduction (64 bins), lane-pair collapse done
//           with V_WMMA_F32_16X16X4_F32 (B = ones -> row sums of A).
//   pass 2: out[f] = P[f] * rsqrt(energy[(f>>4)&63])

#define ANT 64
#define POWER 1.0f

typedef __attribute__((ext_vector_type(2))) float v2f;
typedef __attribute__((ext_vector_type(4))) float f4v;
typedef __attribute__((ext_vector_type(8))) float v8f;

__global__ void zero_energy_kernel(float* __restrict__ g_energy) {
    if (threadIdx.x < ANT) g_energy[threadIdx.x] = 0.0f;
}

// Pass 1: each wave streams contiguous 1KB tiles (256 floats = 16 antenna
// chunks). Lane L (L<16) covers floats [16*L, 16*L+8) of the tile, lane L+16
// covers [16*L+8, 16*L+16) -> lanes L and L+16 share antenna Ab + (L&15).
// Wave stride = gridDim*8*256 floats, a multiple of 1024 (antenna period),
// so each lane's antenna is loop-invariant. After the loop, one WMMA with
// B=ones sums lane pairs (m, m+16) into D row m, replicated across columns:
//   lane 0  holds rows 0..7  in c[0..7]  -> antennas Ab+0 .. Ab+7
//   lane 16 holds rows 8..15 in c[0..7]  -> antennas Ab+8 .. Ab+15
__global__ __launch_bounds__(256) void energy_kernel(const float* __restrict__ P,
                                                     float* __restrict__ g_energy,
                                                     int N) {
    __shared__ float sh[ANT];
    if (threadIdx.x < ANT) sh[threadIdx.x] = 0.0f;
    __syncthreads();

    const int lane      = threadIdx.x & 31;
    const int waveInBlk = threadIdx.x >> 5;
    const long long Wid = (long long)blockIdx.x * 8 + waveInBlk;
    const long long TW  = (long long)gridDim.x * 8;      // total waves, multiple of 8
    const int Ab        = 16 * (int)(Wid & 3);           // wave's antenna base (fixed)
    const int laneOff   = (lane & 15) * 16 + (lane >> 4) * 8;

    float acc = 0.0f;
    for (long long base = Wid * 256; base + 256 <= (long long)N; base += TW * 256) {
        const float* p = P + base + laneOff;
        f4v v0 = *(const f4v*)(p);
        f4v v1 = *(const f4v*)(p + 4);
        acc = fmaf(v0.x, v0.x, acc);
        acc = fmaf(v0.y, v0.y, acc);
        acc = fmaf(v0.z, v0.z, acc);
        acc = fmaf(v0.w, v0.w, acc);
        acc = fmaf(v1.x, v1.x, acc);
        acc = fmaf(v1.y, v1.y, acc);
        acc = fmaf(v1.z, v1.z, acc);
        acc = fmaf(v1.w, v1.w, acc);
    }

    // A[m,0] = acc(lane m), A[m,2] = acc(lane m+16), A[m,1]=A[m,3]=0.
    // B = all-ones (layout independent).  D[m,n] = acc[m] + acc[m+16].
    v2f a; a.x = acc;  a.y = 0.0f;
    v2f b; b.x = 1.0f; b.y = 1.0f;
    v8f c = {0.0f, 0.0f, 0.0f, 0.0f, 0.0f, 0.0f, 0.0f, 0.0f};
    c = __builtin_amdgcn_wmma_f32_16x16x4_f32(
            /*neg_a=*/false, a, /*neg_b=*/false, b,
            /*c_mod=*/(short)0, c, /*reuse_a=*/false, /*reuse_b=*/false);

    if ((lane & 15) == 0) {
        const int rbase = Ab + ((lane >> 4) << 3);
        atomicAdd(&sh[rbase + 0], c[0]);
        atomicAdd(&sh[rbase + 1], c[1]);
        atomicAdd(&sh[rbase + 2], c[2]);
        atomicAdd(&sh[rbase + 3], c[3]);
        atomicAdd(&sh[rbase + 4], c[4]);
        atomicAdd(&sh[rbase + 5], c[5]);
        atomicAdd(&sh[rbase + 6], c[6]);
        atomicAdd(&sh[rbase + 7], c[7]);
    }
    __syncthreads();
    if (threadIdx.x < ANT) atomicAdd(&g_energy[threadIdx.x], sh[threadIdx.x]);
}

// Pass 2: blockDim=256, stride = gridDim*1024 floats (multiple of the 1024-
// float antenna period) -> thread t's antenna is always t>>2; scale computed
// once per thread. Non-temporal stores keep the (L2-resident) input warm.
__global__ __launch_bounds__(256) void scale_kernel(const float* __restrict__ P,
                                                    const float* __restrict__ g_energy,
                                                    float* __restrict__ out,
                                                    int N) {
    const int t  = threadIdx.x;
    const float e = g_energy[t >> 2];
    const float s = __fsqrt_rn(POWER / e);

    long long i            = (long long)blockIdx.x * 1024 + (long long)t * 4;
    const long long stride = (long long)gridDim.x * 1024;
    for (; i + 4 <= (long long)N; i += stride) {
        f4v v = *(const f4v*)(P + i);
        v.x *= s; v.y *= s; v.z *= s; v.w *= s;
        __builtin_nontemporal_store(v, (f4v*)(out + i));
    }
}

extern "C" void kernel_launch(void* const* d_in, const int* in_sizes, int n_in,
                              void* d_out, int out_size, void* d_ws, size_t ws_size,
                              hipStream_t stream) {
    const float* P      = (const float*)d_in[0];
    float*       out    = (float*)d_out;
    float*       energy = (float*)d_ws;       // 64 floats of scratch
    const int    N      = in_sizes[0];        // B * 2*M*K = 67,108,864

    zero_energy_kernel<<<1, 64, 0, stream>>>(energy);
    energy_kernel<<<512, 256, 0, stream>>>(P, energy, N);
    scale_kernel<<<2048, 256, 0, stream>>>(P, energy, out, N);
}